// MUTransformer_944892805635
// MI455X (gfx1250) — compile-verified
//
#include <hip/hip_runtime.h>
#include <math.h>

typedef float v2f __attribute__((ext_vector_type(2)));
typedef float v8f __attribute__((ext_vector_type(8)));
typedef int   v4i __attribute__((ext_vector_type(4)));

#define T_TOK 2048
#define VOCAB 50257
#define RC    256
#define DDIM  1024

#if __has_builtin(__builtin_amdgcn_global_load_async_to_lds_b128) && \
    __has_builtin(__builtin_amdgcn_s_wait_asynccnt)
#define USE_ASYNC_LDS 1
typedef __attribute__((address_space(1))) v4i glb_v4i;   // global int4 (builtin's param type)
typedef __attribute__((address_space(3))) v4i lds_v4i;   // LDS int4
#endif

// ---------------------------------------------------------------------------
// Kernel 1: MU[t][j] = token_to_mu_w[j][ids[t]] + pos[t][j]   (also an output)
// ---------------------------------------------------------------------------
__global__ __launch_bounds__(256) void mu_kernel(const int* __restrict__ ids,
                                                 const float* __restrict__ w_mu,
                                                 const float* __restrict__ pos,
                                                 float* __restrict__ mu_out) {
    const int t = blockIdx.x;        // token
    const int j = threadIdx.x;       // 0..255 == rc
    const int id = ids[t];
    const float v = w_mu[(size_t)j * VOCAB + id] + pos[t * RC + j];
    mu_out[(size_t)t * RC + j] = v;
}

// ---------------------------------------------------------------------------
// Kernel 2: h = gelu(MU @ w1^T + b1)   via V_WMMA_F32_16X16X4_F32
// One wave per 16x16 output tile. 8192 tiles total -> 1024 blocks of 8 waves.
// ---------------------------------------------------------------------------
__global__ __launch_bounds__(256) void gemm1_gelu_kernel(const float* __restrict__ x,
                                                         const float* __restrict__ w1,
                                                         const float* __restrict__ b1,
                                                         float* __restrict__ h) {
    const int wv   = threadIdx.x >> 5;
    const int lane = threadIdx.x & 31;
    const int m    = lane & 15;
    const int grp  = lane >> 4;

    const int tile = blockIdx.x * 8 + wv;   // 0..8191
    const int tt   = tile >> 6;             // token tile 0..127
    const int dt   = tile & 63;             // d-output tile 0..63

    const float* arow = x  + (size_t)(tt * 16 + m) * RC + grp * 2;
    const float* brow = w1 + (size_t)(dt * 16 + m) * RC + grp * 2;

    v8f acc = {0.f, 0.f, 0.f, 0.f, 0.f, 0.f, 0.f, 0.f};
    for (int k = 0; k < RC; k += 4) {
        v2f a = *(const v2f*)(arow + k);
        v2f b = *(const v2f*)(brow + k);
        acc = __builtin_amdgcn_wmma_f32_16x16x4_f32(false, a, false, b,
                                                    (short)0, acc, false, false);
    }

    const int col  = dt * 16 + m;
    const float bias = b1[col];
#pragma unroll
    for (int i = 0; i < 8; ++i) {
        const int row = tt * 16 + i + grp * 8;
        float v = acc[i] + bias;
        v = 0.5f * v * (1.0f + erff(v * 0.70710678118654752f));   // exact gelu
        h[(size_t)row * DDIM + col] = v;
    }
}

// ---------------------------------------------------------------------------
// Kernel 3: row-wise LayerNorm over D=1024
// ---------------------------------------------------------------------------
__global__ __launch_bounds__(256) void ln_kernel(const float* __restrict__ h,
                                                 const float* __restrict__ ln_w,
                                                 const float* __restrict__ ln_b,
                                                 float* __restrict__ hn) {
    const int t   = blockIdx.x;
    const int tid = threadIdx.x;
    __shared__ float s1[256];
    __shared__ float s2[256];

    float vals[4];
    float a = 0.f, b = 0.f;
#pragma unroll
    for (int j = 0; j < 4; ++j) {
        const float v = h[(size_t)t * DDIM + tid + j * 256];
        vals[j] = v;
        a += v;
        b += v * v;
    }
    s1[tid] = a; s2[tid] = b;
    __syncthreads();
    for (int off = 128; off > 0; off >>= 1) {
        if (tid < off) { s1[tid] += s1[tid + off]; s2[tid] += s2[tid + off]; }
        __syncthreads();
    }
    const float mean = s1[0] * (1.0f / DDIM);
    const float var  = s2[0] * (1.0f / DDIM) - mean * mean;
    const float inv  = rsqrtf(var + 1e-6f);
#pragma unroll
    for (int j = 0; j < 4; ++j) {
        const int idx = tid + j * 256;
        hn[(size_t)t * DDIM + idx] = (vals[j] - mean) * inv * ln_w[idx] + ln_b[idx];
    }
}

// ---------------------------------------------------------------------------
// Kernel 4: logits = hn @ w2^T + b2      (the 211-GFLOP, memory-bound GEMM)
// Block: 256 threads / 8 waves, tile 64 tokens x 128 vocab.
// A (hn) staged into LDS per 32-wide K chunk — via GLOBAL_LOAD_ASYNC_TO_LDS
// (direct memory->LDS, ASYNCcnt-tracked) when the builtin exists.
// Each wave: 16-vocab stripe of w2 (global b64 frags), 4 accumulator tiles.
// Logits stored with NT hint so the 412MB stream doesn't evict w2 from L2.
// ---------------------------------------------------------------------------
__global__ __launch_bounds__(256) void gemm2_kernel(const float* __restrict__ hn,
                                                    const float* __restrict__ w2,
                                                    const float* __restrict__ b2,
                                                    float* __restrict__ logits) {
    __shared__ __align__(16) float hs[64 * 36];   // 64 rows x 32 cols, padded (bank-conflict-free)

    const int wv   = threadIdx.x >> 5;
    const int lane = threadIdx.x & 31;
    const int m    = lane & 15;
    const int grp  = lane >> 4;

    const int tb = blockIdx.x * 64;                 // token base
    const int vb = blockIdx.y * 128 + wv * 16;      // this wave's vocab base

    const int vrow = vb + m;
    const int cr   = vrow < VOCAB ? vrow : (VOCAB - 1);   // clamp (EXEC stays full for WMMA)
    const float* bbase = w2 + (size_t)cr * DDIM + grp * 2;

    // cooperative-staging indices: each thread moves 2 x 16B per chunk
    const int r0 = threadIdx.x >> 3;        // 0..31
    const int q4 = (threadIdx.x & 7) * 4;   // 0,4,...,28

    v8f acc0 = {0.f,0.f,0.f,0.f,0.f,0.f,0.f,0.f};
    v8f acc1 = acc0, acc2 = acc0, acc3 = acc0;

    for (int k = 0; k < DDIM; k += 32) {
        __syncthreads();                     // hs free of previous-chunk readers
#ifdef USE_ASYNC_LDS
        __builtin_amdgcn_global_load_async_to_lds_b128(
            (glb_v4i*)&hn[(size_t)(tb + r0) * DDIM + k + q4],
            (lds_v4i*)&hs[r0 * 36 + q4], 0, 0);
        __builtin_amdgcn_global_load_async_to_lds_b128(
            (glb_v4i*)&hn[(size_t)(tb + r0 + 32) * DDIM + k + q4],
            (lds_v4i*)&hs[(r0 + 32) * 36 + q4], 0, 0);
        __builtin_amdgcn_s_wait_asynccnt(0);
#else
        *(float4*)&hs[r0 * 36 + q4] =
            *(const float4*)&hn[(size_t)(tb + r0) * DDIM + k + q4];
        *(float4*)&hs[(r0 + 32) * 36 + q4] =
            *(const float4*)&hn[(size_t)(tb + r0 + 32) * DDIM + k + q4];
#endif
        __syncthreads();

#pragma unroll
        for (int kk = 0; kk < 32; kk += 4) {
            v2f b  = *(const v2f*)(bbase + k + kk);
            v2f a0 = *(const v2f*)&hs[(     m) * 36 + kk + grp * 2];
            v2f a1 = *(const v2f*)&hs[(16 + m) * 36 + kk + grp * 2];
            v2f a2 = *(const v2f*)&hs[(32 + m) * 36 + kk + grp * 2];
            v2f a3 = *(const v2f*)&hs[(48 + m) * 36 + kk + grp * 2];
            acc0 = __builtin_amdgcn_wmma_f32_16x16x4_f32(false, a0, false, b, (short)0, acc0, false, false);
            acc1 = __builtin_amdgcn_wmma_f32_16x16x4_f32(false, a1, false, b, (short)0, acc1, false, false);
            acc2 = __builtin_amdgcn_wmma_f32_16x16x4_f32(false, a2, false, b, (short)0, acc2, false, false);
            acc3 = __builtin_amdgcn_wmma_f32_16x16x4_f32(false, a3, false, b, (short)0, acc3, false, false);
        }
    }

    const int vcol = vb + m;
    if (vcol < VOCAB) {
        const float bias = b2[vcol];
#pragma unroll
        for (int i = 0; i < 8; ++i) {
            const int tok = tb + i + grp * 8;
            __builtin_nontemporal_store(acc0[i] + bias, &logits[(size_t)(tok     ) * VOCAB + vcol]);
            __builtin_nontemporal_store(acc1[i] + bias, &logits[(size_t)(tok + 16) * VOCAB + vcol]);
            __builtin_nontemporal_store(acc2[i] + bias, &logits[(size_t)(tok + 32) * VOCAB + vcol]);
            __builtin_nontemporal_store(acc3[i] + bias, &logits[(size_t)(tok + 48) * VOCAB + vcol]);
        }
    }
}

// ---------------------------------------------------------------------------
extern "C" void kernel_launch(void* const* d_in, const int* in_sizes, int n_in,
                              void* d_out, int out_size, void* d_ws, size_t ws_size,
                              hipStream_t stream) {
    const int*   ids  = (const int*)  d_in[0];
    const float* w_mu = (const float*)d_in[1];
    const float* pos  = (const float*)d_in[2];
    const float* w1   = (const float*)d_in[3];
    const float* b1   = (const float*)d_in[4];
    const float* lnw  = (const float*)d_in[5];
    const float* lnb  = (const float*)d_in[6];
    const float* w2   = (const float*)d_in[7];
    const float* b2   = (const float*)d_in[8];

    float* logits = (float*)d_out;                               // [T, V]
    float* mu     = logits + (size_t)T_TOK * VOCAB;              // [T, rc] (2nd output)
    float* h      = (float*)d_ws;                                // [T, D]
    float* hnorm  = h + (size_t)T_TOK * DDIM;                    // [T, D]

    mu_kernel<<<T_TOK, 256, 0, stream>>>(ids, w_mu, pos, mu);
    gemm1_gelu_kernel<<<(T_TOK / 16) * (DDIM / 16) / 8, 256, 0, stream>>>(mu, w1, b1, h);
    ln_kernel<<<T_TOK, 256, 0, stream>>>(h, lnw, lnb, hnorm);
    dim3 g2(T_TOK / 64, (VOCAB + 127) / 128);                    // token-fastest for L2 reuse of w2
    gemm2_kernel<<<g2, 256, 0, stream>>>(hnorm, w2, b2, logits);
}